// BandSplit_48266842472951
// MI455X (gfx1250) — compile-verified
//
#include <hip/hip_runtime.h>

// ---------------------------------------------------------------------------
// BandSplit: per-band BatchNorm1d(training) + 1x1 Conv1d, folded into a single
// scaled GEMM executed with V_WMMA_F32_16X16X4_F32 on gfx1250 (wave32).
//   in : spec [8, 257, 2, 4000] f32  == [8, 514, 4000]
//   out: [8, 31, 64, 4000] f32
// ---------------------------------------------------------------------------

#define NB     31
#define BATCH  8
#define T      4000
#define CH     514          // total input channels (2*257)
#define CO     64           // conv output channels
#define KPAD   36           // max ci (34) padded to multiple of 4
#define NTILE  250          // T/16
#define NRED   (BATCH * T)  // 32000 elements per channel for BN stats

typedef __attribute__((ext_vector_type(2))) float v2f;
typedef __attribute__((ext_vector_type(8))) float v8f;

__device__ __constant__ int BAND_CI[NB] = {
    4, 6, 6, 6, 6, 6, 6, 6, 6, 6, 6,
    16, 16, 16, 16, 16, 16, 16, 16, 16, 16, 16, 16,
    32, 32, 32, 32, 32, 32, 32, 34};

__device__ __constant__ int BAND_START2[NB] = {
    0, 4, 10, 16, 22, 28, 34, 40, 46, 52, 58,
    64, 80, 96, 112, 128, 144, 160, 176, 192, 208, 224, 240,
    256, 288, 320, 352, 384, 416, 448, 480};

struct BandPtrs {
    const float* w[NB];     // [64, ci]
    const float* bias[NB];  // [64]
    const float* gamma[NB]; // [ci]
    const float* beta[NB];  // [ci]
};

// ---------------------------------------------------------------------------
// Pass 1: per-channel partial sum / sumsq over 2 batches each (4 splits).
// psum/psumsq layout: [CH][4]. Regular (RT) loads pull x into L2 where it
// stays resident for the GEMM pass (66 MB << 192 MB L2).
// ---------------------------------------------------------------------------
__global__ __launch_bounds__(256) void bs_stats_kernel(
    const float* __restrict__ x, float* __restrict__ psum,
    float* __restrict__ psumsq) {
    const int c  = blockIdx.x;       // channel 0..513
    const int bs = blockIdx.y;       // batch split 0..3
    float s = 0.f, q = 0.f;
    #pragma unroll
    for (int bi = 0; bi < 2; ++bi) {
        const int b = bs * 2 + bi;
        const float* p = x + (size_t)(b * CH + c) * T;
        for (int t = threadIdx.x; t < T; t += 256) {
            float v = p[t];
            s += v;
            q += v * v;
        }
    }
    __shared__ float ls[256];
    __shared__ float lq[256];
    ls[threadIdx.x] = s;
    lq[threadIdx.x] = q;
    __syncthreads();
    for (int off = 128; off > 0; off >>= 1) {
        if (threadIdx.x < off) {
            ls[threadIdx.x] += ls[threadIdx.x + off];
            lq[threadIdx.x] += lq[threadIdx.x + off];
        }
        __syncthreads();
    }
    if (threadIdx.x == 0) {
        psum[c * 4 + bs]   = ls[0];
        psumsq[c * 4 + bs] = lq[0];
    }
}

// ---------------------------------------------------------------------------
// Pass 2: fold BN into conv weights.
//   Wf[band][o][k] = W[o][k] * gamma[k] * rsqrt(var+eps)   (0 for k >= ci)
//   bf[band][o]    = bias[o] + sum_k W[o][k]*(beta[k] - mean[k]*scale[k])
// ---------------------------------------------------------------------------
__global__ __launch_bounds__(64) void bs_fold_kernel(
    BandPtrs bp, const float* __restrict__ psum,
    const float* __restrict__ psumsq, float* __restrict__ Wf,
    float* __restrict__ bf) {
    const int band = blockIdx.x;
    const int o    = threadIdx.x;  // 0..63
    const int ci   = BAND_CI[band];
    const int s2   = BAND_START2[band];
    const float* w = bp.w[band];
    float bacc = bp.bias[band][o];
    const float invN = 1.0f / (float)NRED;
    for (int c = 0; c < KPAD; ++c) {
        float wfold = 0.f;
        if (c < ci) {
            const int gc = s2 + c;
            float s = psum[gc * 4 + 0] + psum[gc * 4 + 1] +
                      psum[gc * 4 + 2] + psum[gc * 4 + 3];
            float q = psumsq[gc * 4 + 0] + psumsq[gc * 4 + 1] +
                      psumsq[gc * 4 + 2] + psumsq[gc * 4 + 3];
            float mean  = s * invN;
            float var   = q * invN - mean * mean;
            float scale = bp.gamma[band][c] * __frsqrt_rn(var + 1e-5f);
            float wv    = w[o * ci + c];
            wfold = wv * scale;
            bacc += wv * (bp.beta[band][c] - mean * scale);
        }
        Wf[(size_t)(band * CO + o) * KPAD + c] = wfold;
    }
    bf[band * CO + o] = bacc;
}

// ---------------------------------------------------------------------------
// Pass 3: y[b, band, o, t] = sum_k Wf[band][o][k] * x[b, s2+k, t] + bf
// One wave per 16-wide time tile, computing all 4 M-tiles (64 outputs) with
// V_WMMA_F32_16X16X4_F32.
//   - A fragments (weights) and bias fragments preloaded once per wave.
//   - B-fragment element offsets are tile-invariant: precomputed as 32-bit
//     voffsets so per-tile loads are saddr+voffset with a single add.
//   - All B loads for a tile are issued before the WMMA block (one wait,
//     then up to 36 back-to-back WMMAs).
//   - First kstep consumes the bias fragment directly as the C operand.
//   - Output stores are non-temporal so the 254 MB of write-once output
//     does not evict the L2-resident input.
// All WMMA-guarding conditions are wave-uniform => EXEC stays all-ones.
// ---------------------------------------------------------------------------
__global__ __launch_bounds__(256) void bs_gemm_kernel(
    const float* __restrict__ x, const float* __restrict__ Wf,
    const float* __restrict__ bf, float* __restrict__ out) {
    const int band  = blockIdx.z;
    const int bb    = blockIdx.y;
    const int chunk = blockIdx.x;            // 0..3 (64 tiles per chunk)
    const int wave  = threadIdx.x >> 5;      // 0..7
    const int lane  = threadIdx.x & 31;
    const int lo    = lane & 15;             // column / M-row selector
    const int hi    = lane >> 4;             // lane-half: K/M offset select

    const int ci     = BAND_CI[band];
    const int start2 = BAND_START2[band];
    const int ksteps = (ci + 3) >> 2;        // 1..9, wave-uniform

    // ---- Preload A fragments (folded weights) into registers -------------
    // A 16x4 f32 layout: lanes 0-15 rows M, v0 holds K=k (lo half)/K=k+2 (hi),
    // v1 holds K=k+1 / K=k+3.  Wf row-major [o][KPAD] makes this a b64 load.
    v2f afrag[9][4];
    #pragma unroll
    for (int ks = 0; ks < 9; ++ks) {
        if (ks < ksteps) {
            #pragma unroll
            for (int mt = 0; mt < 4; ++mt) {
                const int m = mt * 16 + lo;
                const float* p =
                    Wf + (size_t)(band * CO + m) * KPAD + ks * 4 + hi * 2;
                afrag[ks][mt].x = p[0];
                afrag[ks][mt].y = p[1];
            }
        }
    }

    // ---- Bias fragments (C-matrix operand of the first kstep) ------------
    // C/D layout: VGPR r holds row M=r (lanes 0-15) / M=8+r (lanes 16-31).
    v8f binit[4];
    #pragma unroll
    for (int mt = 0; mt < 4; ++mt) {
        const float* p = bf + band * CO + mt * 16 + hi * 8;
        #pragma unroll
        for (int r = 0; r < 8; ++r) binit[mt][r] = p[r];
    }

    // ---- Tile-invariant 32-bit B-load offsets (elements) ------------------
    // B 4x16 f32: lanes 0-15 -> K=k,k+1 ; lanes 16-31 -> K=k+2,k+3.
    unsigned boff0[9], boff1[9];
    #pragma unroll
    for (int ks = 0; ks < 9; ++ks) {
        if (ks < ksteps) {
            const int k0 = ks * 4 + hi * 2;
            int ch0 = start2 + k0;
            int ch1 = ch0 + 1;
            if (ch0 > CH - 1) ch0 = CH - 1;  // padded-K cols have W'=0
            if (ch1 > CH - 1) ch1 = CH - 1;
            boff0[ks] = (unsigned)(ch0 * T + lo);
            boff1[ks] = (unsigned)(ch1 * T + lo);
        }
    }

    const float* xb = x + (size_t)bb * CH * T;
    float* ob = out + (size_t)(bb * NB + band) * CO * T;

    for (int i = 0; i < 8; ++i) {
        const int tile = chunk * 64 + i * 8 + wave;  // wave-uniform
        if (tile >= NTILE) break;
        const unsigned t0 = (unsigned)(tile * 16);

        // ---- Issue all B-fragment loads for this tile up front -----------
        v2f bfr[9];
        #pragma unroll
        for (int ks = 0; ks < 9; ++ks) {
            if (ks < ksteps) {
                bfr[ks].x = xb[boff0[ks] + t0];
                bfr[ks].y = xb[boff1[ks] + t0];
            }
        }

        // ---- WMMA chain: kstep 0 consumes bias fragment as C --------------
        v8f acc[4];
        #pragma unroll
        for (int mt = 0; mt < 4; ++mt) {
            acc[mt] = __builtin_amdgcn_wmma_f32_16x16x4_f32(
                false, afrag[0][mt], false, bfr[0], (short)0, binit[mt],
                false, false);
        }
        #pragma unroll
        for (int ks = 1; ks < 9; ++ks) {
            if (ks < ksteps) {
                #pragma unroll
                for (int mt = 0; mt < 4; ++mt) {
                    acc[mt] = __builtin_amdgcn_wmma_f32_16x16x4_f32(
                        false, afrag[ks][mt], false, bfr[ks], (short)0,
                        acc[mt], false, false);
                }
            }
        }

        // ---- Non-temporal stores: row (mt*16 + hi*8 + r), col t0+lo -------
        #pragma unroll
        for (int mt = 0; mt < 4; ++mt) {
            #pragma unroll
            for (int r = 0; r < 8; ++r) {
                const int m = mt * 16 + hi * 8 + r;
                __builtin_nontemporal_store(acc[mt][r],
                                            &ob[(size_t)m * T + t0 + lo]);
            }
        }
    }
}

// ---------------------------------------------------------------------------
extern "C" void kernel_launch(void* const* d_in, const int* in_sizes, int n_in,
                              void* d_out, int out_size, void* d_ws,
                              size_t ws_size, hipStream_t stream) {
    const float* x = (const float*)d_in[0];
    float* ws = (float*)d_ws;

    // workspace layout (floats)
    float* psum   = ws;                       // [514*4]
    float* psumsq = psum + CH * 4;            // [514*4]
    float* Wf     = psumsq + CH * 4;          // [31*64*36]
    float* bf     = Wf + NB * CO * KPAD;      // [31*64]
    float* out    = (float*)d_out;

    BandPtrs bp;
    for (int i = 0; i < NB; ++i) {
        bp.w[i]     = (const float*)d_in[1 + i];
        bp.bias[i]  = (const float*)d_in[1 + NB + i];
        bp.gamma[i] = (const float*)d_in[1 + 2 * NB + i];
        bp.beta[i]  = (const float*)d_in[1 + 3 * NB + i];
    }

    bs_stats_kernel<<<dim3(CH, 4), 256, 0, stream>>>(x, psum, psumsq);
    bs_fold_kernel<<<NB, CO, 0, stream>>>(bp, psum, psumsq, Wf, bf);
    bs_gemm_kernel<<<dim3(4, BATCH, NB), 256, 0, stream>>>(x, Wf, bf, out);
}